// Encoder_36601711296974
// MI455X (gfx1250) — compile-verified
//
#include <hip/hip_runtime.h>
#include <hip/hip_bf16.h>

typedef float v2f __attribute__((ext_vector_type(2)));
typedef float v8f __attribute__((ext_vector_type(8)));

#define DF 256   // feature dim (K and N of every GEMM)
#define MT 64    // rows per block
#define KC 32    // K chunk staged in LDS

// EPI: 0 = bias only, 1 = bias+relu, 2 = bias+relu+row L2 normalize
// DUAL: accumulate A0@W0 + A1@W1
template<int EPI, bool DUAL>
__global__ __launch_bounds__(256) void gemm_wmma_kernel(
    const float* __restrict__ A0, const float* __restrict__ A1,
    const float* __restrict__ W0, const float* __restrict__ W1,
    const float* __restrict__ bias, float* __restrict__ Out, int nrows)
{
  // A staged transposed: Ast[k][row]; both arrays padded so the two 16-lane
  // half-wave groups (K=k0/k0+1 vs K=k0+2/k0+3) hit disjoint LDS banks.
  __shared__ float Ast[KC][MT + 8];
  __shared__ float Bs[KC][DF + 8];
  __shared__ float rowsum[MT];

  const int tid   = threadIdx.x;
  const int lane  = tid & 31;
  const int wave  = tid >> 5;
  const int msel  = wave & 1;   // which pair of 16-row slabs (rows msel*32..)
  const int nquad = wave >> 1;  // which 64-col quad (0..3)
  const int row_base = blockIdx.x * MT;

  if (EPI == 2 && tid < MT) rowsum[tid] = 0.0f;

  v8f c[2][4];
#pragma unroll
  for (int s = 0; s < 2; ++s)
#pragma unroll
    for (int t = 0; t < 4; ++t)
#pragma unroll
      for (int j = 0; j < 8; ++j) c[s][t][j] = 0.0f;

  const int m    = lane & 15;          // row-in-slab / col-in-tile
  const int koff = (lane >> 4) << 1;   // lanes 0-15: K=k0,k0+1 ; lanes 16-31: K=k0+2,k0+3

  for (int pass = 0; pass < (DUAL ? 2 : 1); ++pass) {
    const float* __restrict__ A = (DUAL && pass) ? A1 : A0;
    const float* __restrict__ W = (DUAL && pass) ? W1 : W0;
    for (int kc = 0; kc < DF; kc += KC) {
      __syncthreads();
      // stage A chunk (64x32) transposed: 512 float4 slots, 2 per thread
#pragma unroll
      for (int i = 0; i < 2; ++i) {
        int slot = tid + i * 256;
        int r  = slot >> 3;          // row within tile
        int c4 = (slot & 7) << 2;    // k offset within chunk
        float4 v = make_float4(0.f, 0.f, 0.f, 0.f);
        if (row_base + r < nrows)
          v = *(const float4*)(A + (size_t)(row_base + r) * DF + kc + c4);
        Ast[c4 + 0][r] = v.x;
        Ast[c4 + 1][r] = v.y;
        Ast[c4 + 2][r] = v.z;
        Ast[c4 + 3][r] = v.w;
      }
      // stage B chunk (32x256): 2048 float4 slots, 8 per thread
#pragma unroll
      for (int i = 0; i < 8; ++i) {
        int slot = tid + i * 256;
        int r  = slot >> 6;
        int c4 = (slot & 63) << 2;
        *(float4*)&Bs[r][c4] = *(const float4*)(W + (size_t)(kc + r) * DF + c4);
      }
      __syncthreads();
#pragma unroll
      for (int kk = 0; kk < KC; kk += 4) {
        // two A fragments (slabs), four B fragments (col tiles):
        // 6 DS loads feeding 8 WMMAs, A reused 4x, B reused 2x.
        v2f a[2];
#pragma unroll
        for (int s = 0; s < 2; ++s) {
          int row = (msel * 2 + s) * 16 + m;
          a[s].x = Ast[kk + koff + 0][row];
          a[s].y = Ast[kk + koff + 1][row];
        }
#pragma unroll
        for (int t = 0; t < 4; ++t) {
          int col = nquad * 64 + t * 16 + m;
          v2f b;
          b.x = Bs[kk + koff + 0][col];
          b.y = Bs[kk + koff + 1][col];
#pragma unroll
          for (int s = 0; s < 2; ++s) {
            c[s][t] = __builtin_amdgcn_wmma_f32_16x16x4_f32(
                false, a[s], false, b, (short)0, c[s][t], false, false);
          }
        }
      }
    }
  }

  // ---- epilogue ----
  // C layout: VGPR j -> row (lane<16 ? j : j+8) of the slab, col = lane&15
  float sq[2][8];
#pragma unroll
  for (int s = 0; s < 2; ++s)
#pragma unroll
    for (int j = 0; j < 8; ++j) sq[s][j] = 0.0f;
#pragma unroll
  for (int t = 0; t < 4; ++t) {
    int col = nquad * 64 + t * 16 + m;
    float bv = bias[col];
#pragma unroll
    for (int s = 0; s < 2; ++s) {
#pragma unroll
      for (int j = 0; j < 8; ++j) {
        float v = c[s][t][j] + bv;
        if (EPI >= 1) v = fmaxf(v, 0.0f);
        c[s][t][j] = v;
        if (EPI == 2) sq[s][j] += v * v;
      }
    }
  }
  if (EPI == 2) {
#pragma unroll
    for (int s = 0; s < 2; ++s) {
      int rl = (msel * 2 + s) * 16 + ((lane < 16) ? 0 : 8);
#pragma unroll
      for (int j = 0; j < 8; ++j) atomicAdd(&rowsum[rl + j], sq[s][j]);
    }
    __syncthreads();
  }
#pragma unroll
  for (int s = 0; s < 2; ++s) {
    int rl = (msel * 2 + s) * 16 + ((lane < 16) ? 0 : 8);
#pragma unroll
    for (int j = 0; j < 8; ++j) {
      int grow = row_base + rl + j;
      if (grow >= nrows) continue;
      float scale = 1.0f;
      if (EPI == 2) {
        float nrm = sqrtf(rowsum[rl + j]);
        scale = 1.0f / fmaxf(nrm, 1e-12f);
      }
#pragma unroll
      for (int t = 0; t < 4; ++t) {
        int col = nquad * 64 + t * 16 + m;
        Out[(size_t)grow * DF + col] = c[s][t][j] * scale;
      }
    }
  }
}

// segment-max scatter: 64 threads per edge, float4 rows, int-bit atomic max
// (valid because messages are relu outputs, i.e. >= 0, and agg is zeroed:
//  zero-in-degree nodes get 0 exactly as the reference's isfinite() fixup).
__global__ __launch_bounds__(256) void scatter_max_kernel(
    const float* __restrict__ msg, const int* __restrict__ src,
    const int* __restrict__ dst, float* __restrict__ agg, int E)
{
  int e = blockIdx.x * 4 + (threadIdx.x >> 6);
  if (e >= E) return;
  int l = (threadIdx.x & 63) << 2;
  int s = src[e];
  int d = dst[e];
  float4 v = *(const float4*)(msg + (size_t)s * DF + l);
  int* o = (int*)(agg + (size_t)d * DF + l);
  atomicMax(o + 0, __float_as_int(v.x));
  atomicMax(o + 1, __float_as_int(v.y));
  atomicMax(o + 2, __float_as_int(v.z));
  atomicMax(o + 3, __float_as_int(v.w));
}

extern "C" void kernel_launch(void* const* d_in, const int* in_sizes, int n_in,
                              void* d_out, int out_size, void* d_ws, size_t ws_size,
                              hipStream_t stream) {
  (void)n_in; (void)out_size; (void)ws_size;
  const float* x   = (const float*)d_in[0];
  const int*   src = (const int*)d_in[1];
  const int*   dst = (const int*)d_in[2];
  const float* Wp0 = (const float*)d_in[3];
  const float* bp0 = (const float*)d_in[4];
  const float* Ws0 = (const float*)d_in[5];
  const float* Wn0 = (const float*)d_in[6];
  const float* b0  = (const float*)d_in[7];
  const float* Wp1 = (const float*)d_in[8];
  const float* bp1 = (const float*)d_in[9];
  const float* Ws1 = (const float*)d_in[10];
  const float* Wn1 = (const float*)d_in[11];
  const float* b1  = (const float*)d_in[12];

  const int N = in_sizes[0] / DF;
  const int E = in_sizes[1];
  const size_t ND = (size_t)N * DF;

  float* m_buf = (float*)d_ws;       // pooled messages  [N,256]
  float* agg   = m_buf + ND;         // segment-max accumulator [N,256]
  float* outF  = (float*)d_out;      // final h          [N,256]
  float* enc   = outF + ND;          // enc_feat_input   [N,256]

  dim3 blk(256);
  int gblocks = (N + MT - 1) / MT;
  int sblocks = (E + 3) / 4;

  // ---- layer 0 ----
  gemm_wmma_kernel<1, false><<<gblocks, blk, 0, stream>>>(x, nullptr, Wp0, nullptr, bp0, m_buf, N);
  hipMemsetAsync(agg, 0, ND * sizeof(float), stream);
  scatter_max_kernel<<<sblocks, blk, 0, stream>>>(m_buf, src, dst, agg, E);
  gemm_wmma_kernel<2, true><<<gblocks, blk, 0, stream>>>(x, agg, Ws0, Wn0, b0, enc, N);

  // ---- layer 1 ----
  gemm_wmma_kernel<1, false><<<gblocks, blk, 0, stream>>>(enc, nullptr, Wp1, nullptr, bp1, m_buf, N);
  hipMemsetAsync(agg, 0, ND * sizeof(float), stream);
  scatter_max_kernel<<<sblocks, blk, 0, stream>>>(m_buf, src, dst, agg, E);
  gemm_wmma_kernel<0, true><<<gblocks, blk, 0, stream>>>(enc, agg, Ws1, Wn1, b1, outF, N);
}